// Full_Bridge_11201274707992
// MI455X (gfx1250) — compile-verified
//
#include <hip/hip_runtime.h>
#include <hip/hip_bf16.h>

typedef __attribute__((ext_vector_type(2))) float v2f;
typedef __attribute__((ext_vector_type(8))) float v8f;

#define DEV_INLINE __device__ __forceinline__

DEV_INLINE float sigmoidf_(float x) { return 1.0f / (1.0f + __expf(-x)); }
DEV_INLINE float siluf_(float x)    { return x * sigmoidf_(x); }
DEV_INLINE float softplusf_(float x){ return (x > 20.0f) ? x : log1pf(__expf(x)); }

// ---------------------------------------------------------------------------
// Gather: build c0 (B, 28928, 8) from t1..t5 channel slices (group0 sequences
// c1|c2|c3|c4 concatenated along L), and c1b (B, 64, 16) for group1.
// ---------------------------------------------------------------------------
__global__ void gather0_k(const float* __restrict__ t1, const float* __restrict__ t2,
                          const float* __restrict__ t3, const float* __restrict__ t4,
                          const float* __restrict__ t5, float* __restrict__ c0, int total) {
  int idx = blockIdx.x * blockDim.x + threadIdx.x;
  if (idx >= total) return;
  int ch = idx & 7;
  int rl = idx >> 3;
  int l  = rl % 28928;
  int b  = rl / 28928;
  const float* src; int C, HW, chg, ll;
  if      (l < 16384) { src=t1; C=8;  HW=16384; chg=ch;    ll=l;        }
  else if (l < 20480) { src=t2; C=16; HW=4096;  chg=ch;    ll=l-16384;  }
  else if (l < 21504) { src=t3; C=24; HW=1024;  chg=ch;    ll=l-20480;  }
  else if (l < 21760) { src=t4; C=32; HW=256;   chg=ch;    ll=l-21504;  }
  else if (l < 21824) { src=t5; C=48; HW=64;    chg=ch;    ll=l-21760;  }
  else if (l < 25920) { src=t2; C=16; HW=4096;  chg=ch+8;  ll=l-21824;  }
  else if (l < 26944) { src=t3; C=24; HW=1024;  chg=ch+8;  ll=l-25920;  }
  else if (l < 27200) { src=t4; C=32; HW=256;   chg=ch+8;  ll=l-26944;  }
  else if (l < 27264) { src=t5; C=48; HW=64;    chg=ch+8;  ll=l-27200;  }
  else if (l < 28288) { src=t3; C=24; HW=1024;  chg=ch+16; ll=l-27264;  }
  else if (l < 28544) { src=t4; C=32; HW=256;   chg=ch+16; ll=l-28288;  }
  else if (l < 28608) { src=t5; C=48; HW=64;    chg=ch+16; ll=l-28544;  }
  else if (l < 28864) { src=t4; C=32; HW=256;   chg=ch+24; ll=l-28608;  }
  else                { src=t5; C=48; HW=64;    chg=ch+24; ll=l-28864;  }
  c0[idx] = src[((size_t)b * C + chg) * HW + ll];
}

__global__ void gather1_k(const float* __restrict__ t5, float* __restrict__ c1b, int total) {
  int idx = blockIdx.x * blockDim.x + threadIdx.x;
  if (idx >= total) return;
  int ch = idx & 15;
  int rl = idx >> 4;
  int l  = rl & 63;
  int b  = rl >> 6;
  c1b[idx] = t5[((size_t)b * 48 + 32 + ch) * 64 + l];
}

// ---------------------------------------------------------------------------
// Generic per-token GEMM with WMMA: out[row, j] = sum_k in[row, k] * W[j, k]
// A tile = 16 tokens x DIN (f32), done with V_WMMA_F32_16X16X4_F32:
//   A 16x4 layout : lane t (m=t&15, half=t>>4): a.x=A[m][4ks+2h], a.y=A[m][4ks+2h+1]
//   B 4x16 layout : lane t (n=t&15, half=t>>4): b.x=B[4ks+2h][n], b.y=B[4ks+2h+1][n]
//   C/D 16x16     : VGPR r, lane t: D[r + 8*half][n]
// NOUT columns are valid (weights masked to zero beyond); NSTORE = row stride.
// One wave per 16-token tile; 8 waves per block. Tile guard is wave-uniform so
// EXEC stays all-ones across the WMMA (ISA requirement).
// ---------------------------------------------------------------------------
template <int DIN, int NOUT, int NSTORE>
__global__ void tokgemm_k(const float* __restrict__ in, const float* __restrict__ W,
                          float* __restrict__ out, int nrows) {
  constexpr int KS = DIN / 4;
  constexpr int NT = (NOUT + 15) / 16;
  int wave = threadIdx.x >> 5;
  int lane = threadIdx.x & 31;
  int tile = blockIdx.x * 8 + wave;
  if (tile * 16 >= nrows) return;              // wave-uniform exit
  int m    = lane & 15;
  int half = lane >> 4;
  int row0 = tile * 16;

  v2f a[KS];
  const float* inrow = in + (size_t)(row0 + m) * DIN;
#pragma unroll
  for (int ks = 0; ks < KS; ++ks) {
    int kb = ks * 4 + 2 * half;
    a[ks].x = inrow[kb];
    a[ks].y = inrow[kb + 1];
  }

#pragma unroll
  for (int nt = 0; nt < NT; ++nt) {
    int j = nt * 16 + m;                        // B operand column (= lane&15)
    v8f acc = {};
#pragma unroll
    for (int ks = 0; ks < KS; ++ks) {
      int kb = ks * 4 + 2 * half;
      v2f bf;
      bf.x = (j < NOUT) ? W[(size_t)j * DIN + kb]     : 0.0f;
      bf.y = (j < NOUT) ? W[(size_t)j * DIN + kb + 1] : 0.0f;
      acc = __builtin_amdgcn_wmma_f32_16x16x4_f32(false, a[ks], false, bf,
                                                  (short)0, acc, false, false);
    }
    if (j < NOUT) {
#pragma unroll
      for (int r = 0; r < 8; ++r) {
        int mm = r + 8 * half;
        out[(size_t)(row0 + mm) * NSTORE + j] = acc[r];
      }
    }
  }
}

// ---------------------------------------------------------------------------
// Depthwise causal conv (K=4) + SiLU. xi = first DI columns of xz (stride 2*DI).
// Zero-pad at each sequence start (starts.x..w are sequence start offsets).
// ---------------------------------------------------------------------------
template <int DI>
__global__ void conv_silu_k(const float* __restrict__ xz, const float* __restrict__ cw,
                            const float* __restrict__ cb, float* __restrict__ xc,
                            int Ltot, int4 starts, int total) {
  int idx = blockIdx.x * blockDim.x + threadIdx.x;
  if (idx >= total) return;
  int d  = idx % DI;
  int rl = idx / DI;          // b*Ltot + l
  int l  = rl % Ltot;
  int ls = (l >= starts.w) ? starts.w
         : (l >= starts.z) ? starts.z
         : (l >= starts.y) ? starts.y : starts.x;
  float acc = cb[d];
#pragma unroll
  for (int k = 0; k < 4; ++k) {
    int lp  = l - 3 + k;
    float v = (lp >= ls) ? xz[(size_t)(rl - 3 + k) * (2 * DI) + d] : 0.0f;
    acc += cw[d * 4 + k] * v;
  }
  xc[(size_t)rl * DI + d] = siluf_(acc);
}

// delta[row, d] = softplus(dt[row] * dt_w[d] + dt_b[d]);  dt = xdbl[row, 0]
template <int DI>
__global__ void delta_k(const float* __restrict__ xdbl, const float* __restrict__ dtw,
                        const float* __restrict__ dtb, float* __restrict__ delta, int total) {
  int idx = blockIdx.x * blockDim.x + threadIdx.x;
  if (idx >= total) return;
  int d   = idx % DI;
  int row = idx / DI;
  float x = xdbl[(size_t)row * 33] * dtw[d] + dtb[d];
  delta[idx] = softplusf_(x);
}

// ---------------------------------------------------------------------------
// Chunked parallel scan (3 phases). State element per thread: (d, n), ds=16.
// Phase A: per-chunk (prod(a), partial h with h_in=0).
// ---------------------------------------------------------------------------
template <int DI>
__global__ void scan_partial_k(const float* __restrict__ delta, const float* __restrict__ xdbl,
                               const float* __restrict__ xc, const float* __restrict__ Alog,
                               float* __restrict__ aprod, float* __restrict__ bacc,
                               int Ltot, int seqoff, int Lseq, int nch, int cb, int nchtot, int LC) {
  constexpr int DS = 16;
  constexpr int S  = DI * DS;
  int b  = blockIdx.x / nch;
  int ci = blockIdx.x % nch;
  int t  = threadIdx.x;
  int d  = t >> 4, n = t & 15;
  float A = -__expf(Alog[d * DS + n]);
  int l0 = seqoff + ci * LC;
  int l1 = l0 + LC; int lend = seqoff + Lseq; if (l1 > lend) l1 = lend;
  float ap = 1.0f, bc = 0.0f;
  size_t rowbase = (size_t)b * Ltot;
  for (int l = l0; l < l1; ++l) {
    size_t row = rowbase + l;
    float dl = delta[row * DI + d];
    float Bm = xdbl[row * 33 + 1 + n];
    float x  = xc[row * DI + d];
    float a  = __expf(dl * A);
    bc = a * bc + dl * Bm * x;
    ap *= a;
    if (l + 8 < l1) {                         // hide HBM/L2 latency on serial path
      __builtin_prefetch(&delta[(row + 8) * DI + d], 0, 3);
      __builtin_prefetch(&xdbl[(row + 8) * 33 + n], 0, 3);
    }
  }
  size_t g = ((size_t)b * nchtot + cb + ci) * S + t;
  aprod[g] = ap;
  bacc[g]  = bc;
}

// Phase B: sequential scan across chunk states -> per-chunk incoming h.
template <int DI>
__global__ void scan_combine_k(const float* __restrict__ aprod, const float* __restrict__ bacc,
                               float* __restrict__ hin, int nch, int cb, int nchtot) {
  constexpr int S = DI * 16;
  int b = blockIdx.x;
  int t = threadIdx.x;
  float h = 0.0f;
  for (int ci = 0; ci < nch; ++ci) {
    size_t g = ((size_t)b * nchtot + cb + ci) * S + t;
    hin[g] = h;
    h = aprod[g] * h + bacc[g];
  }
}

// Phase C: replay each chunk from its incoming state; y = <h, C> reduced over
// the 16 n-lanes (wave32 half-wave) with shfl_xor; gate with SiLU(z).
template <int DI>
__global__ void scan_apply_k(const float* __restrict__ delta, const float* __restrict__ xdbl,
                             const float* __restrict__ xc, const float* __restrict__ xz,
                             const float* __restrict__ Alog, const float* __restrict__ Dp,
                             const float* __restrict__ hin, float* __restrict__ yg,
                             int Ltot, int seqoff, int Lseq, int nch, int cb, int nchtot, int LC) {
  constexpr int DS = 16;
  constexpr int S  = DI * DS;
  int b  = blockIdx.x / nch;
  int ci = blockIdx.x % nch;
  int t  = threadIdx.x;
  int d  = t >> 4, n = t & 15;
  float A  = -__expf(Alog[d * DS + n]);
  float Dd = Dp[d];
  int l0 = seqoff + ci * LC;
  int l1 = l0 + LC; int lend = seqoff + Lseq; if (l1 > lend) l1 = lend;
  float h = hin[((size_t)b * nchtot + cb + ci) * S + t];
  size_t rowbase = (size_t)b * Ltot;
  for (int l = l0; l < l1; ++l) {
    size_t row = rowbase + l;
    float dl = delta[row * DI + d];
    float Bm = xdbl[row * 33 + 1 + n];
    float Cm = xdbl[row * 33 + 17 + n];
    float x  = xc[row * DI + d];
    float a  = __expf(dl * A);
    h = a * h + dl * Bm * x;
    float p = h * Cm;
    p += __shfl_xor(p, 1);
    p += __shfl_xor(p, 2);
    p += __shfl_xor(p, 4);
    p += __shfl_xor(p, 8);
    if (n == 0) {
      float y = p + x * Dd;
      float z = xz[row * (2 * DI) + DI + d];
      yg[row * DI + d] = y * siluf_(z);
    }
    if (l + 8 < l1)
      __builtin_prefetch(&xdbl[(row + 8) * 33 + n], 0, 3);
  }
}

// ---------------------------------------------------------------------------
// skip-add + LayerNorm(DM) + projection (DM x DM) + bias, per token.
// ---------------------------------------------------------------------------
template <int DM>
__global__ void lnproj_k(const float* __restrict__ mo, const float* __restrict__ c,
                         const float* __restrict__ g, const float* __restrict__ be,
                         const float* __restrict__ pw, const float* __restrict__ pb,
                         const float* __restrict__ skip, float* __restrict__ fin, int rows) {
  int row = blockIdx.x * blockDim.x + threadIdx.x;
  if (row >= rows) return;
  float sk = skip[0];
  float tv[DM];
  float mu = 0.0f;
#pragma unroll
  for (int i = 0; i < DM; ++i) {
    tv[i] = mo[(size_t)row * DM + i] + sk * c[(size_t)row * DM + i];
    mu += tv[i];
  }
  mu *= (1.0f / DM);
  float var = 0.0f;
#pragma unroll
  for (int i = 0; i < DM; ++i) { float e = tv[i] - mu; var += e * e; }
  var *= (1.0f / DM);
  float inv = rsqrtf(var + 1e-5f);
  float ln[DM];
#pragma unroll
  for (int i = 0; i < DM; ++i) ln[i] = (tv[i] - mu) * inv * g[i] + be[i];
#pragma unroll
  for (int j = 0; j < DM; ++j) {
    float acc = pb[j];
#pragma unroll
    for (int i = 0; i < DM; ++i) acc += ln[i] * pw[j * DM + i];
    fin[(size_t)row * DM + j] = acc;
  }
}

// ---------------------------------------------------------------------------
// Scatter fin0/fin1 rows into x1..x5 (B, C, H*W) output layout.
// ---------------------------------------------------------------------------
__global__ void scatter_k(const float* __restrict__ fin0, const float* __restrict__ fin1,
                          float* __restrict__ out, int total) {
  int idx = blockIdx.x * blockDim.x + threadIdx.x;
  if (idx >= total) return;
  const int X1 = 1048576, X2 = 524288, X3 = 196608, X4 = 65536;
  float v;
  if (idx < X1) {                       // x1: (8, 8, 16384)
    int l = idx % 16384; int bc = idx / 16384; int ch = bc % 8; int b = bc / 8;
    v = fin0[((size_t)b * 28928 + l) * 8 + ch];
  } else if (idx < X1 + X2) {           // x2: (8, 16, 4096)
    int r = idx - X1; int l = r % 4096; int bc = r / 4096; int ch = bc % 16; int b = bc / 16;
    if (ch < 8) v = fin0[((size_t)b * 28928 + 16384 + l) * 8 + ch];
    else        v = fin0[((size_t)b * 28928 + 21824 + l) * 8 + (ch - 8)];
  } else if (idx < X1 + X2 + X3) {      // x3: (8, 24, 1024)
    int r = idx - X1 - X2; int l = r % 1024; int bc = r / 1024; int ch = bc % 24; int b = bc / 24;
    if      (ch < 8)  v = fin0[((size_t)b * 28928 + 20480 + l) * 8 + ch];
    else if (ch < 16) v = fin0[((size_t)b * 28928 + 25920 + l) * 8 + (ch - 8)];
    else              v = fin0[((size_t)b * 28928 + 27264 + l) * 8 + (ch - 16)];
  } else if (idx < X1 + X2 + X3 + X4) { // x4: (8, 32, 256)
    int r = idx - X1 - X2 - X3; int l = r % 256; int bc = r / 256; int ch = bc % 32; int b = bc / 32;
    if      (ch < 8)  v = fin0[((size_t)b * 28928 + 21504 + l) * 8 + ch];
    else if (ch < 16) v = fin0[((size_t)b * 28928 + 26944 + l) * 8 + (ch - 8)];
    else if (ch < 24) v = fin0[((size_t)b * 28928 + 28288 + l) * 8 + (ch - 16)];
    else              v = fin0[((size_t)b * 28928 + 28608 + l) * 8 + (ch - 24)];
  } else {                              // x5: (8, 48, 64)
    int r = idx - X1 - X2 - X3 - X4; int l = r % 64; int bc = r / 64; int ch = bc % 48; int b = bc / 48;
    if      (ch < 8)  v = fin0[((size_t)b * 28928 + 21760 + l) * 8 + ch];
    else if (ch < 16) v = fin0[((size_t)b * 28928 + 27200 + l) * 8 + (ch - 8)];
    else if (ch < 24) v = fin0[((size_t)b * 28928 + 28544 + l) * 8 + (ch - 16)];
    else if (ch < 32) v = fin0[((size_t)b * 28928 + 28864 + l) * 8 + (ch - 24)];
    else              v = fin1[((size_t)b * 64 + l) * 16 + (ch - 32)];
  }
  out[idx] = v;
}

// ---------------------------------------------------------------------------
// Host launcher
// ---------------------------------------------------------------------------
extern "C" void kernel_launch(void* const* d_in, const int* in_sizes, int n_in,
                              void* d_out, int out_size, void* d_ws, size_t ws_size,
                              hipStream_t stream) {
  const float* t1 = (const float*)d_in[0];
  const float* t2 = (const float*)d_in[1];
  const float* t3 = (const float*)d_in[2];
  const float* t4 = (const float*)d_in[3];
  const float* t5 = (const float*)d_in[4];
  const float* m0_in_w   = (const float*)d_in[5];
  const float* m0_conv_w = (const float*)d_in[6];
  const float* m0_conv_b = (const float*)d_in[7];
  const float* m0_xproj  = (const float*)d_in[8];
  const float* m0_dt_w   = (const float*)d_in[9];
  const float* m0_dt_b   = (const float*)d_in[10];
  const float* m0_Alog   = (const float*)d_in[11];
  const float* m0_D      = (const float*)d_in[12];
  const float* m0_out_w  = (const float*)d_in[13];
  const float* m1_in_w   = (const float*)d_in[14];
  const float* m1_conv_w = (const float*)d_in[15];
  const float* m1_conv_b = (const float*)d_in[16];
  const float* m1_xproj  = (const float*)d_in[17];
  const float* m1_dt_w   = (const float*)d_in[18];
  const float* m1_dt_b   = (const float*)d_in[19];
  const float* m1_Alog   = (const float*)d_in[20];
  const float* m1_D      = (const float*)d_in[21];
  const float* m1_out_w  = (const float*)d_in[22];
  const float* norm_g  = (const float*)d_in[23];
  const float* norm_b  = (const float*)d_in[24];
  const float* proj_w  = (const float*)d_in[25];
  const float* proj_b  = (const float*)d_in[26];
  const float* norm1_g = (const float*)d_in[27];
  const float* norm1_b = (const float*)d_in[28];
  const float* proj1_w = (const float*)d_in[29];
  const float* proj1_b = (const float*)d_in[30];
  const float* skip    = (const float*)d_in[31];
  float* out = (float*)d_out;

  const int B = 8;
  const int L0 = 28928;                 // 21824 + 5440 + 1344 + 320
  const size_t R0 = (size_t)B * L0;     // 231424 rows (multiple of 16)
  const int L1 = 64;
  const size_t R1 = (size_t)B * L1;     // 512 rows

  // sequence metadata (group0)
  const int seqoff[4] = {0, 21824, 27264, 28608};
  const int seqlen[4] = {21824, 5440, 1344, 320};
  const int LC = 128;
  const int nch[4]    = {171, 43, 11, 3};
  const int cbase[4]  = {0, 171, 214, 225};
  const int NCHTOT0   = 228;

  // workspace carve (floats)
  float* p = (float*)d_ws;
  float* c0     = p; p += R0 * 8;
  float* xz0    = p; p += R0 * 32;
  float* xc0    = p; p += R0 * 16;
  float* xdbl0  = p; p += R0 * 33;
  float* delta0 = p; p += R0 * 16;
  float* yg0    = p; p += R0 * 16;
  float* mo0    = p; p += R0 * 8;
  float* fin0   = p; p += R0 * 8;
  float* ap0    = p; p += (size_t)B * NCHTOT0 * 256;
  float* bc0    = p; p += (size_t)B * NCHTOT0 * 256;
  float* hi0    = p; p += (size_t)B * NCHTOT0 * 256;
  float* c1b    = p; p += R1 * 16;
  float* xz1    = p; p += R1 * 64;
  float* xc1    = p; p += R1 * 32;
  float* xdbl1  = p; p += R1 * 33;
  float* delta1 = p; p += R1 * 32;
  float* yg1    = p; p += R1 * 32;
  float* mo1    = p; p += R1 * 16;
  float* fin1   = p; p += R1 * 16;
  float* ap1    = p; p += (size_t)B * 512;
  float* bc1    = p; p += (size_t)B * 512;
  float* hi1    = p; p += (size_t)B * 512;

  const int TB = 256;
  auto grids = [](size_t n, int b) { return (int)((n + b - 1) / b); };

  // 1) gather
  gather0_k<<<grids(R0 * 8, TB), TB, 0, stream>>>(t1, t2, t3, t4, t5, c0, (int)(R0 * 8));
  gather1_k<<<grids(R1 * 16, TB), TB, 0, stream>>>(t5, c1b, (int)(R1 * 16));

  // 2) in-projection (WMMA): xz = c @ in_w^T
  tokgemm_k<8, 32, 32><<<grids(R0 / 16, 8), TB, 0, stream>>>(c0, m0_in_w, xz0, (int)R0);
  tokgemm_k<16, 64, 64><<<grids(R1 / 16, 8), TB, 0, stream>>>(c1b, m1_in_w, xz1, (int)R1);

  // 3) depthwise conv (K=4) + SiLU on xi half of xz
  conv_silu_k<16><<<grids(R0 * 16, TB), TB, 0, stream>>>(
      xz0, m0_conv_w, m0_conv_b, xc0, L0, make_int4(0, 21824, 27264, 28608), (int)(R0 * 16));
  conv_silu_k<32><<<grids(R1 * 32, TB), TB, 0, stream>>>(
      xz1, m1_conv_w, m1_conv_b, xc1, L1, make_int4(0, 1 << 30, 1 << 30, 1 << 30), (int)(R1 * 32));

  // 4) x-projection (WMMA): xdbl = xc @ xproj_w^T  (NOUT=33 -> dt|B|C)
  tokgemm_k<16, 33, 33><<<grids(R0 / 16, 8), TB, 0, stream>>>(xc0, m0_xproj, xdbl0, (int)R0);
  tokgemm_k<32, 33, 33><<<grids(R1 / 16, 8), TB, 0, stream>>>(xc1, m1_xproj, xdbl1, (int)R1);

  // 5) delta = softplus(dt * dt_w + dt_b)
  delta_k<16><<<grids(R0 * 16, TB), TB, 0, stream>>>(xdbl0, m0_dt_w, m0_dt_b, delta0, (int)(R0 * 16));
  delta_k<32><<<grids(R1 * 32, TB), TB, 0, stream>>>(xdbl1, m1_dt_w, m1_dt_b, delta1, (int)(R1 * 32));

  // 6) chunked parallel scan, group0 (state 16x16 = 256 threads/block)
  for (int s = 0; s < 4; ++s) {
    scan_partial_k<16><<<B * nch[s], 256, 0, stream>>>(
        delta0, xdbl0, xc0, m0_Alog, ap0, bc0,
        L0, seqoff[s], seqlen[s], nch[s], cbase[s], NCHTOT0, LC);
    scan_combine_k<16><<<B, 256, 0, stream>>>(ap0, bc0, hi0, nch[s], cbase[s], NCHTOT0);
    scan_apply_k<16><<<B * nch[s], 256, 0, stream>>>(
        delta0, xdbl0, xc0, xz0, m0_Alog, m0_D, hi0, yg0,
        L0, seqoff[s], seqlen[s], nch[s], cbase[s], NCHTOT0, LC);
  }
  // group1 (state 32x16 = 512 threads/block, single chunk)
  scan_partial_k<32><<<B, 512, 0, stream>>>(delta1, xdbl1, xc1, m1_Alog, ap1, bc1,
                                            L1, 0, L1, 1, 0, 1, L1);
  scan_combine_k<32><<<B, 512, 0, stream>>>(ap1, bc1, hi1, 1, 0, 1);
  scan_apply_k<32><<<B, 512, 0, stream>>>(delta1, xdbl1, xc1, xz1, m1_Alog, m1_D, hi1, yg1,
                                          L1, 0, L1, 1, 0, 1, L1);

  // 7) out-projection (WMMA): mo = yg @ out_w^T
  tokgemm_k<16, 8, 8><<<grids(R0 / 16, 8), TB, 0, stream>>>(yg0, m0_out_w, mo0, (int)R0);
  tokgemm_k<32, 16, 16><<<grids(R1 / 16, 8), TB, 0, stream>>>(yg1, m1_out_w, mo1, (int)R1);

  // 8) skip + layernorm + projection
  lnproj_k<8><<<grids(R0, TB), TB, 0, stream>>>(mo0, c0, norm_g, norm_b, proj_w, proj_b,
                                                skip, fin0, (int)R0);
  lnproj_k<16><<<grids(R1, TB), TB, 0, stream>>>(mo1, c1b, norm1_g, norm1_b, proj1_w, proj1_b,
                                                 skip, fin1, (int)R1);

  // 9) scatter into x1..x5
  scatter_k<<<grids((size_t)out_size, TB), TB, 0, stream>>>(fin0, fin1, out, out_size);
}